// Aggregator_27642409517733
// MI455X (gfx1250) — compile-verified
//
#include <hip/hip_runtime.h>

typedef __attribute__((ext_vector_type(2))) float v2f;
typedef __attribute__((ext_vector_type(8))) float v8f;

#define EMB 64
#define LDS_STRIDE 68  // 64 + 4 pad: row r starts at bank (4*r)%64 -> 16 distinct banks

// Order-preserving monotonic map float -> uint so we can use integer atomicMax.
__device__ __forceinline__ unsigned enc_f32(float f) {
  unsigned u = __float_as_uint(f);
  return (u & 0x80000000u) ? ~u : (u | 0x80000000u);
}
__device__ __forceinline__ float dec_f32(unsigned e) {
  unsigned u = (e & 0x80000000u) ? (e & 0x7fffffffu) : ~e;
  return __uint_as_float(u);
}

// ---------------------------------------------------------------- K0: init
__global__ void k_init(float* __restrict__ out, unsigned* __restrict__ segmax,
                       float* __restrict__ segsum, int N) {
  int i = blockIdx.x * blockDim.x + threadIdx.x;
  if (i < N * EMB) out[i] = 0.0f;
  if (i < N) { segmax[i] = 0u; segsum[i] = 0.0f; }  // 0u < enc of any real float
}

// ---------------------------------------------------------------- K1: scores via WMMA
// Each wave handles 16 edges. Stage HR = head*rel and TAIL rows (16x64 f32 each)
// into LDS, then D = HR x TAIL^T with v_wmma_f32_16x16x4_f32 accumulated over
// K=64 (16 chunks of K=4). score_e = D[e][e].
__global__ __launch_bounds__(128)
void k_scores(const float* __restrict__ entity, const int* __restrict__ head,
              const int* __restrict__ tail, const int* __restrict__ etype,
              const float* __restrict__ rel, float* __restrict__ scores,
              unsigned* __restrict__ segmax, int E) {
  __shared__ float lds[4 * 2 * 16 * LDS_STRIDE];  // 4 waves x (HR + TAIL)
  const int lane = threadIdx.x & 31;
  const int wave = threadIdx.x >> 5;
  float* hrL = lds + wave * (2 * 16 * LDS_STRIDE);
  float* tlL = hrL + 16 * LDS_STRIDE;

  const int eBase = blockIdx.x * 64 + wave * 16;
  // every lane mirrors edge (lane&15)'s metadata so shuffles are sourced from
  // always-active lanes
  int e = eBase + (lane & 15);
  if (e >= E) e = E - 1;  // clamp padding (writes are guarded later)
  const int hid = head[e];
  const int tid_ = tail[e];
  const int rt = etype[e];

  // ---- stage: 2 rows per iteration, 16 lanes x float4 per row
#pragma unroll
  for (int i = 0; i < 8; ++i) {
    int m = 2 * i + (lane >> 4);  // edge row 0..15
    int col = (lane & 15) * 4;    // 0..60
    int hm = __shfl(hid, m, 32);
    int tm = __shfl(tid_, m, 32);
    int rm = __shfl(rt, m, 32);
    float4 hv = *(const float4*)(entity + (size_t)hm * EMB + col);
    float4 rv = *(const float4*)(rel + (size_t)rm * EMB + col);
    float4 tv = *(const float4*)(entity + (size_t)tm * EMB + col);
    float4 hr;
    hr.x = hv.x * rv.x; hr.y = hv.y * rv.y;
    hr.z = hv.z * rv.z; hr.w = hv.w * rv.w;
    *(float4*)(hrL + m * LDS_STRIDE + col) = hr;
    *(float4*)(tlL + m * LDS_STRIDE + col) = tv;
  }
  __syncthreads();

  // ---- compute: A (16x4) lane L<16 holds M=L,K={k0,k0+1}; L>=16 holds K={k0+2,k0+3}.
  // B (4x16) mirrors: v0/v1 lanes0-15 = rows k0,k0+1; lanes16-31 = rows k0+2,k0+3,
  // i.e. lane L supplies TAIL[edge L&15][same k pair] -> identical per-lane mapping.
  v8f acc = {0.f, 0.f, 0.f, 0.f, 0.f, 0.f, 0.f, 0.f};
  const int row = lane & 15;
  const int sel = (lane >> 4) * 2;
  const float* aP = hrL + row * LDS_STRIDE + sel;
  const float* bP = tlL + row * LDS_STRIDE + sel;
#pragma unroll
  for (int c = 0; c < 16; ++c) {
    v2f a = *(const v2f*)(aP + 4 * c);
    v2f b = *(const v2f*)(bP + 4 * c);
    acc = __builtin_amdgcn_wmma_f32_16x16x4_f32(false, a, false, b,
                                                (short)0, acc, false, false);
  }

  // ---- diagonal extraction:
  // D 16x16 f32 layout: lanes0-15: N=lane, M=vgpr; lanes16-31: N=lane-16, M=vgpr+8.
  // diag d<8 -> lane d, vgpr d ; diag d>=8 -> lane 16+d, vgpr d-8.
  const bool lowerOk = (lane < 8);
  const bool upperOk = (lane >= 24);
  const int d = (lowerOk ? lane : (lane - 16)) & 15;
  const int g = (lowerOk ? lane : (lane - 24)) & 7;
  float s = acc[0];
#pragma unroll
  for (int gg = 1; gg < 8; ++gg)
    if (g == gg) s = acc[gg];
  const int hD = __shfl(hid, d, 32);  // shuffle while all lanes active
  const int eD = eBase + d;
  if ((lowerOk || upperOk) && eD < E) {
    scores[eD] = s;
    atomicMax(segmax + hD, enc_f32(s));
  }
}

// ---------------------------------------------------------------- K2: exp + segment sum
__global__ void k_softmax_den(const int* __restrict__ head, float* sc /* in: scores, out: ex */,
                              const unsigned* __restrict__ segmax,
                              float* __restrict__ segsum, int E) {
  int i = blockIdx.x * blockDim.x + threadIdx.x;
  if (i < E) {
    int h = head[i];
    float x = expf(sc[i] - dec_f32(segmax[h]));
    sc[i] = x;
    atomicAdd(segsum + h, x);
  }
}

// ---------------------------------------------------------------- K3: weighted scatter-add
// 16 lanes (one half-wave) per edge, float4 per lane.
__global__ void k_aggregate(const float* __restrict__ entity, const int* __restrict__ head,
                            const int* __restrict__ tail, const float* __restrict__ ex,
                            const float* __restrict__ segsum, float* __restrict__ out, int E) {
  long long gid = (long long)blockIdx.x * blockDim.x + threadIdx.x;
  int e = (int)(gid >> 4);
  int col = ((int)gid & 15) * 4;
  if (e < E) {
    int h = head[e];
    int t = tail[e];
    float attn = ex[e] / segsum[h];
    float4 tv = *(const float4*)(entity + (size_t)t * EMB + col);
    float* o = out + (size_t)h * EMB + col;
    atomicAdd(o + 0, attn * tv.x);
    atomicAdd(o + 1, attn * tv.y);
    atomicAdd(o + 2, attn * tv.z);
    atomicAdd(o + 3, attn * tv.w);
  }
}

// ---------------------------------------------------------------- launcher
extern "C" void kernel_launch(void* const* d_in, const int* in_sizes, int n_in,
                              void* d_out, int out_size, void* d_ws, size_t ws_size,
                              hipStream_t stream) {
  const float* entity = (const float*)d_in[0];
  const int* edge_index = (const int*)d_in[1];
  const int* etype = (const int*)d_in[2];
  const float* rel = (const float*)d_in[3];

  const int N = in_sizes[0] / EMB;   // 80000
  const int E = in_sizes[2];         // 1280000 (edge_type length)
  const int* head = edge_index;      // edge_index[0][:]
  const int* tail = edge_index + E;  // edge_index[1][:]

  // workspace: [scores/ex : E floats][segmax : N u32][segsum : N floats]
  float* sc = (float*)d_ws;
  unsigned* segmax = (unsigned*)(sc + E);
  float* segsum = (float*)(segmax + N);
  float* out = (float*)d_out;

  const int initElems = N * EMB;
  k_init<<<(initElems + 255) / 256, 256, 0, stream>>>(out, segmax, segsum, N);

  k_scores<<<(E + 63) / 64, 128, 0, stream>>>(entity, head, tail, etype, rel,
                                              sc, segmax, E);

  k_softmax_den<<<(E + 255) / 256, 256, 0, stream>>>(head, sc, segmax, segsum, E);

  const long long aggThreads = (long long)E * 16;
  k_aggregate<<<(int)((aggThreads + 255) / 256), 256, 0, stream>>>(
      entity, head, tail, sc, segsum, out, E);
}